// CombinedLoss_37881611550866
// MI455X (gfx1250) — compile-verified
//
#include <hip/hip_runtime.h>
#include <hip/hip_bf16.h>

typedef __attribute__((ext_vector_type(2))) float v2f;
typedef __attribute__((ext_vector_type(4))) float v4f;
typedef __attribute__((ext_vector_type(8))) float v8f;

static constexpr int kCls  = 13;   // classification columns
static constexpr int kCols = 16;   // 3 regression + 13 classification

// Focal-BCE term, restructured:
//   z = (t==1) ? -x : x
//   ce   = softplus(z)  = log(1+exp(-|x|)) + max(z,0)
//   1-pt = sigmoid(z)   = (z>=0) ? 1/(1+e) : e/(1+e),  e = exp(-|x|)
//   wgt  = (t==1) ? bw : 1-bw ; zero where t == -1
__device__ __forceinline__ float focal_term(float x, float t,
                                            float w1, float w0) {
    float e    = __expf(-__builtin_fabsf(x));       // v_exp_f32
    float ope  = 1.0f + e;
    float L    = __logf(ope);                       // v_log_f32 path
    bool  pos  = (t == 1.0f);
    float z    = pos ? -x : x;
    float ce   = L + fmaxf(z, 0.0f);
    float inv  = __builtin_amdgcn_rcpf(ope);        // raw v_rcp_f32
    float ompt = (z >= 0.0f) ? inv : e * inv;       // sigmoid(z) = 1 - p_t
    float wgt  = pos ? w1 : w0;
    float v    = ce * ompt * ompt * wgt;            // gamma = 2
    return (t == -1.0f) ? 0.0f : v;
}

__global__ __launch_bounds__(256)
void loss_partial_kernel(const v4f* __restrict__ out4,
                         const v4f* __restrict__ tgt4,
                         const float* __restrict__ bw,
                         float*       __restrict__ partials,
                         int nrows, float invB) {
    // Class weights + complements held in registers.
    float w1[kCls], w0[kCls];
#pragma unroll
    for (int c = 0; c < kCls; ++c) {
        float b = bw[c];
        w1[c] = b;
        w0[c] = 1.0f - b;
    }

    const float rw0  = 10.0f * invB;   // fold mean + 10x weight
    const float rw12 = invB;

    float acc = 0.0f;
    const int stride = gridDim.x * blockDim.x;
    for (int r = blockIdx.x * blockDim.x + threadIdx.x; r < nrows; r += stride) {
        // One row = 64 contiguous bytes per stream; streamed once -> NT hint.
        v4f o[4], t[4];
#pragma unroll
        for (int q = 0; q < 4; ++q) {
            o[q] = __builtin_nontemporal_load(&out4[r * 4 + q]);
            t[q] = __builtin_nontemporal_load(&tgt4[r * 4 + q]);
        }
        float of[kCols], tf[kCols];
#pragma unroll
        for (int q = 0; q < 4; ++q) {
#pragma unroll
            for (int j = 0; j < 4; ++j) {
                of[q * 4 + j] = o[q][j];
                tf[q * 4 + j] = t[q][j];
            }
        }
        // Regression: 10*mse0 + mse1 + mse2, weights pre-divided by B.
        float d0 = of[0] - tf[0];
        float d1 = of[1] - tf[1];
        float d2 = of[2] - tf[2];
        acc += rw0 * d0 * d0 + rw12 * d1 * d1 + rw12 * d2 * d2;
        // Focal BCE over 13 classification columns.
#pragma unroll
        for (int c = 0; c < kCls; ++c)
            acc += focal_term(of[3 + c], tf[3 + c], w1[c], w0[c]);
    }

    // ---- Block reduction: LDS fold to 32 lanes, then WMMA row-sum trick ----
    __shared__ float sdata[256];
    sdata[threadIdx.x] = acc;
    __syncthreads();
    if (threadIdx.x < 32) {            // wave 0, EXEC all ones (uniform branch)
        float s = 0.0f;
#pragma unroll
        for (int i = 0; i < 8; ++i) s += sdata[threadIdx.x + i * 32];
        // D = A x ones(4x16): every D column holds the 16 row-sums of A,
        // where A packs the 32 lane partials (16x4 f32, second half zero).
        v2f a; a.x = s;    a.y = 0.0f;
        v2f b; b.x = 1.0f; b.y = 1.0f;  // all-ones B regardless of layout
        v8f cz = {0.0f, 0.0f, 0.0f, 0.0f, 0.0f, 0.0f, 0.0f, 0.0f};
        v8f d = __builtin_amdgcn_wmma_f32_16x16x4_f32(
            /*neg_a=*/false, a, /*neg_b=*/false, b,
            /*c_mod=*/(short)0, cz, /*reuse_a=*/false, /*reuse_b=*/false);
        float rsum = d[0] + d[1] + d[2] + d[3] + d[4] + d[5] + d[6] + d[7];
        rsum += __shfl_xor(rsum, 16, 32);
        if (threadIdx.x == 0) partials[blockIdx.x] = rsum;
    }
}

// Final reduction of block partials in fp64 (tiny; one block).
__global__ __launch_bounds__(256)
void finalize_kernel(const float* __restrict__ partials, int n,
                     float* __restrict__ out) {
    __shared__ double sd[256];
    double s = 0.0;
    for (int i = threadIdx.x; i < n; i += 256) s += (double)partials[i];
    sd[threadIdx.x] = s;
    __syncthreads();
    for (int off = 128; off > 0; off >>= 1) {
        if ((int)threadIdx.x < off) sd[threadIdx.x] += sd[threadIdx.x + off];
        __syncthreads();
    }
    if (threadIdx.x == 0) out[0] = (float)sd[0];
}

extern "C" void kernel_launch(void* const* d_in, const int* in_sizes, int n_in,
                              void* d_out, int out_size, void* d_ws, size_t ws_size,
                              hipStream_t stream) {
    const float* out_p = (const float*)d_in[0];   // (B, 16) float32
    const float* tgt_p = (const float*)d_in[1];   // (B, 16) float32
    const float* bw_p  = (const float*)d_in[2];   // (13,)   float32

    const int total = in_sizes[0];
    const int nrows = total / kCols;              // B = 2,097,152

    const int threads = 256;
    int blocks = (nrows + threads - 1) / threads;
    if (blocks > 4096) blocks = 4096;             // each thread: ~2 rows
    const int max_ws_blocks = (int)(ws_size / sizeof(float));
    if (blocks > max_ws_blocks) blocks = max_ws_blocks;
    if (blocks < 1) blocks = 1;

    const float invB = 1.0f / (float)nrows;

    loss_partial_kernel<<<blocks, threads, 0, stream>>>(
        (const v4f*)out_p, (const v4f*)tgt_p, bw_p,
        (float*)d_ws, nrows, invB);
    finalize_kernel<<<1, 256, 0, stream>>>((const float*)d_ws, blocks,
                                           (float*)d_out);
}